// MRConv2d_v2_85615878078951
// MI455X (gfx1250) — compile-verified
//
#include <hip/hip_runtime.h>
#include <math.h>

typedef __attribute__((ext_vector_type(2))) float v2f;
typedef __attribute__((ext_vector_type(8))) float v8f;

#define C_IN   128
#define TWO_C  256
#define COUT   256
#define KNN    16
#define MT     128          // rows (nodes) per block tile
#define HSTR   260          // padded LDS row stride in floats (bank-conflict avoidance)
#define NB     16384        // N
#define NBATCH 8

__device__ __forceinline__ float4 f4max(float4 a, float4 b) {
    return make_float4(fmaxf(a.x, b.x), fmaxf(a.y, b.y), fmaxf(a.z, b.z), fmaxf(a.w, b.w));
}

// ---------------------------------------------------------------------------
// Kernel A: fused gather+max aggregation -> h tile in LDS -> WMMA f32 GEMM.
// Writes pre-BN y (= h @ W^T + b) into d_out.
// ---------------------------------------------------------------------------
__global__ __launch_bounds__(256) void fused_agg_gemm(
    const float* __restrict__ x,          // (B, N, C_IN)
    const long long* __restrict__ edge,   // (B, N, KNN) int64
    const float* __restrict__ W,          // (COUT, TWO_C) row-major
    const float* __restrict__ bias,       // (COUT)
    float* __restrict__ y)                // (B*N, COUT)
{
    __shared__ float hT[MT][HSTR];        // 128 x 260 floats = 130 KB LDS

    const int tile = blockIdx.x;          // 0 .. B*N/MT - 1  (MT divides N)
    const int m0g  = tile * MT;           // global row base
    const int b    = m0g / NB;            // batch index
    const int nbase = m0g - b * NB;       // node base within batch

    const float*     xb = x    + (size_t)b * NB * C_IN;
    const long long* eb = edge + ((size_t)b * NB + nbase) * KNN;

    const int wid  = threadIdx.x >> 5;    // wave id 0..7
    const int lane = threadIdx.x & 31;

    // ---- Step 1: build h = [x, max_k(neigh) - x] for 128 nodes into LDS ----
    // One wave per node; lane covers 4 channels (32 lanes * 4 = 128).
    for (int r = wid; r < MT; r += 8) {
        const float4* xr = (const float4*)(xb + (size_t)(nbase + r) * C_IN);
        float4 xv = xr[lane];
        float4 mx = make_float4(-3.402823466e38f, -3.402823466e38f,
                                -3.402823466e38f, -3.402823466e38f);
        const long long* er = eb + (size_t)r * KNN;
        #pragma unroll
        for (int k = 0; k < KNN; ++k) {
            long long idx = er[k];
            const float4* nr = (const float4*)(xb + (size_t)idx * C_IN);
            mx = f4max(mx, nr[lane]);
        }
        float4 ag = make_float4(mx.x - xv.x, mx.y - xv.y, mx.z - xv.z, mx.w - xv.w);
        float* hr = hT[r];
        *(float4*)&hr[lane * 4]        = xv;
        *(float4*)&hr[C_IN + lane * 4] = ag;
    }
    __syncthreads();

    // ---- Step 2: y[128 x 256] = h[128 x 256] @ W^T via V_WMMA_F32_16X16X4 ----
    // Wave `wid` owns row tile rows [wid*16, wid*16+16). 16 column tiles,
    // processed 4 accumulators at a time so the A fragment is reused 4x.
    const int l16 = lane & 15;
    const int hi  = lane >> 4;            // 0: K={k,k+1}, 1: K={k+2,k+3}
    const float* hrow = &hT[wid * 16 + l16][2 * hi];

    for (int cg = 0; cg < 4; ++cg) {
        v8f acc0 = {}, acc1 = {}, acc2 = {}, acc3 = {};
        const int n0 = cg * 64;           // first column of this group of 4 tiles
        const float* wp0 = W + (size_t)(n0 +  0 + l16) * TWO_C + 2 * hi;
        const float* wp1 = W + (size_t)(n0 + 16 + l16) * TWO_C + 2 * hi;
        const float* wp2 = W + (size_t)(n0 + 32 + l16) * TWO_C + 2 * hi;
        const float* wp3 = W + (size_t)(n0 + 48 + l16) * TWO_C + 2 * hi;

        #pragma unroll 8
        for (int k = 0; k < TWO_C; k += 4) {
            v2f a  = *(const v2f*)(hrow + k);
            v2f b0 = *(const v2f*)(wp0 + k);
            v2f b1 = *(const v2f*)(wp1 + k);
            v2f b2 = *(const v2f*)(wp2 + k);
            v2f b3 = *(const v2f*)(wp3 + k);
            acc0 = __builtin_amdgcn_wmma_f32_16x16x4_f32(false, a, false, b0, (short)0, acc0, false, false);
            acc1 = __builtin_amdgcn_wmma_f32_16x16x4_f32(false, a, false, b1, (short)0, acc1, false, false);
            acc2 = __builtin_amdgcn_wmma_f32_16x16x4_f32(false, a, false, b2, (short)0, acc2, false, false);
            acc3 = __builtin_amdgcn_wmma_f32_16x16x4_f32(false, a, false, b3, (short)0, acc3, false, false);
        }

        // Store: component r of acc -> row m0g + wid*16 + r + 8*hi, col n0 + j*16 + l16
        const int rowg = m0g + wid * 16 + hi * 8;
        v8f accs[4] = {acc0, acc1, acc2, acc3};
        #pragma unroll
        for (int j = 0; j < 4; ++j) {
            const int col = n0 + j * 16 + l16;
            const float bc = bias[col];
            #pragma unroll
            for (int r = 0; r < 8; ++r) {
                y[(size_t)(rowg + r) * COUT + col] = accs[j][r] + bc;
            }
        }
    }
}

// ---------------------------------------------------------------------------
// Kernel B1: deterministic partial column sums / sums of squares.
// Block i handles rows [i*rowsPer, (i+1)*rowsPer), thread t = column t.
// ---------------------------------------------------------------------------
__global__ __launch_bounds__(256) void colstats_partial(
    const float* __restrict__ y, float* __restrict__ psum, float* __restrict__ psq,
    int rowsPer)
{
    const int t  = threadIdx.x;           // column 0..255
    const size_t r0 = (size_t)blockIdx.x * rowsPer;
    float s = 0.f, s2 = 0.f;
    for (int r = 0; r < rowsPer; ++r) {
        float v = y[(r0 + r) * COUT + t];
        s  += v;
        s2 += v * v;
    }
    psum[blockIdx.x * COUT + t] = s;
    psq [blockIdx.x * COUT + t] = s2;
}

// ---------------------------------------------------------------------------
// Kernel B2: finalize mean/var -> per-column scale/shift (fixed order = deterministic).
// ---------------------------------------------------------------------------
__global__ __launch_bounds__(256) void colstats_final(
    const float* __restrict__ psum, const float* __restrict__ psq,
    const float* __restrict__ gamma, const float* __restrict__ beta,
    float* __restrict__ scale, float* __restrict__ shift,
    int nparts, float invM)
{
    const int t = threadIdx.x;            // column 0..255
    float s = 0.f, s2 = 0.f;
    for (int i = 0; i < nparts; ++i) {
        s  += psum[i * COUT + t];
        s2 += psq [i * COUT + t];
    }
    float mean = s * invM;
    float var  = s2 * invM - mean * mean;
    float sc   = gamma[t] * rsqrtf(var + 1e-5f);
    scale[t] = sc;
    shift[t] = beta[t] - mean * sc;
}

// ---------------------------------------------------------------------------
// Kernel C: in-place BN + exact GELU (erf), vectorized float4.
// ---------------------------------------------------------------------------
__device__ __forceinline__ float gelu_exact(float u) {
    return 0.5f * u * (1.0f + erff(u * 0.70710678118654752f));
}

__global__ __launch_bounds__(256) void bn_gelu(
    float* __restrict__ y, const float* __restrict__ scale, const float* __restrict__ shift)
{
    const size_t i = (size_t)blockIdx.x * blockDim.x + threadIdx.x;  // one float4 each
    const int c4 = ((int)(i & 63)) * 4;   // 256 cols / 4 = 64 float4 per row
    float4 v = ((const float4*)y)[i];
    v.x = gelu_exact(v.x * scale[c4 + 0] + shift[c4 + 0]);
    v.y = gelu_exact(v.y * scale[c4 + 1] + shift[c4 + 1]);
    v.z = gelu_exact(v.z * scale[c4 + 2] + shift[c4 + 2]);
    v.w = gelu_exact(v.w * scale[c4 + 3] + shift[c4 + 3]);
    ((float4*)y)[i] = v;
}

// ---------------------------------------------------------------------------
extern "C" void kernel_launch(void* const* d_in, const int* in_sizes, int n_in,
                              void* d_out, int out_size, void* d_ws, size_t ws_size,
                              hipStream_t stream)
{
    const float*     x     = (const float*)d_in[0];
    const long long* edge  = (const long long*)d_in[1];
    const float*     W     = (const float*)d_in[2];
    const float*     bias  = (const float*)d_in[3];
    const float*     gamma = (const float*)d_in[4];
    const float*     beta  = (const float*)d_in[5];
    float*           y     = (float*)d_out;           // y staged in-place in d_out

    const int rowsTotal = NBATCH * NB;                // 131072
    const int tiles     = rowsTotal / MT;             // 1024

    // Workspace layout: psum[256*256] | psq[256*256] | scale[256] | shift[256]
    float* psum  = (float*)d_ws;
    float* psq   = psum + 256 * COUT;
    float* scale = psq  + 256 * COUT;
    float* shift = scale + COUT;

    fused_agg_gemm<<<tiles, 256, 0, stream>>>(x, edge, W, bias, y);

    const int nparts  = 256;
    const int rowsPer = rowsTotal / nparts;           // 512
    colstats_partial<<<nparts, 256, 0, stream>>>(y, psum, psq, rowsPer);
    colstats_final<<<1, 256, 0, stream>>>(psum, psq, gamma, beta, scale, shift,
                                          nparts, 1.0f / (float)rowsTotal);

    const size_t total4 = (size_t)rowsTotal * COUT / 4;   // 8,388,608
    bn_gelu<<<(int)(total4 / 256), 256, 0, stream>>>(y, scale, shift);
}